// Attention_5403068858414
// MI455X (gfx1250) — compile-verified
//
#include <hip/hip_runtime.h>
#include <hip/hip_bf16.h>

typedef __attribute__((ext_vector_type(2))) float v2f;
typedef __attribute__((ext_vector_type(4))) float v4f;
typedef __attribute__((ext_vector_type(8))) float v8f;

#define N_FEAT 128   // fixed by the reference

// ---------------------------------------------------------------------------
// Kernel 1: Q = x @ W^T  via V_WMMA_F32_16X16X4_F32
//   grid.x  = node tiles (16 rows each)
//   block   = 256 threads = 8 waves; wave w computes output-feature tile w
//   K loop  = 128 / 4 = 32 WMMAs per tile
// ---------------------------------------------------------------------------
__global__ __launch_bounds__(256) void qgemm_wmma_f32(
    const float* __restrict__ x,   // [n_nodes, 128]
    const float* __restrict__ W,   // [128, 128] (out, in)
    float* __restrict__ Q,         // [n_nodes, 128]
    int n_nodes) {
  const int m0   = blockIdx.x << 4;          // node-tile base row
  const int wave = threadIdx.x >> 5;         // 0..7 -> o-tile
  const int lane = threadIdx.x & 31;
  const int o0   = wave << 4;
  const int half = lane >> 4;                // 0: lanes 0-15, 1: lanes 16-31
  const int l15  = lane & 15;

  // clamp row for safe loads on a (non-existent here) remainder tile
  int mrow = m0 + l15;
  if (mrow >= n_nodes) mrow = n_nodes - 1;

  // A-frag source: x[mrow][kb + 2*half + {0,1}]       (contiguous float2)
  const float* arow = x + (size_t)mrow * N_FEAT + 2 * half;
  // B-frag source: W[o0 + l15][kb + 2*half + {0,1}]   (B = W^T tile)
  const float* brow = W + (size_t)(o0 + l15) * N_FEAT + 2 * half;

  v8f c = {};
#pragma unroll
  for (int kb = 0; kb < N_FEAT; kb += 4) {
    v2f a = *(const v2f*)(arow + kb);
    v2f b = *(const v2f*)(brow + kb);
    // (neg_a, A, neg_b, B, c_mod, C, reuse_a, reuse_b)
    c = __builtin_amdgcn_wmma_f32_16x16x4_f32(false, a, false, b,
                                              (short)0, c, false, false);
  }

  // D layout: VGPR r -> (M = r + 8*half, N = l15)
  float* qout = Q + (size_t)m0 * N_FEAT + o0;
#pragma unroll
  for (int r = 0; r < 8; ++r) {
    const int m = r + 8 * half;
    if (m0 + m < n_nodes) {
      qout[(size_t)m * N_FEAT + l15] = c[r];
    }
  }
}

// ---------------------------------------------------------------------------
// Kernel 2: per-edge dot product, one wave32 per edge.
//   Each lane: one coalesced float4 from Q[src] and Q[dst] (128 dims / 32
//   lanes = 4), 4 FMAs, then 5-step shfl_xor reduction. L2-resident gathers.
// ---------------------------------------------------------------------------
__global__ __launch_bounds__(256) void edge_dot(
    const float* __restrict__ Q,          // [n_nodes, 128]
    const long long* __restrict__ ei,     // [2, n_edges] int64
    float* __restrict__ out,              // [n_edges]
    long long n_edges) {
  const int lane = threadIdx.x & 31;
  const int wave = threadIdx.x >> 5;
  const long long e = (long long)blockIdx.x * 8 + wave;
  if (e >= n_edges) return;

  const long long s = ei[e];
  const long long d = ei[n_edges + e];

  v4f a = ((const v4f*)(Q + (size_t)s * N_FEAT))[lane];
  v4f b = ((const v4f*)(Q + (size_t)d * N_FEAT))[lane];
  float p = a.x * b.x + a.y * b.y + a.z * b.z + a.w * b.w;

#pragma unroll
  for (int off = 16; off >= 1; off >>= 1)
    p += __shfl_xor(p, off, 32);

  if (lane == 0) out[e] = p;
}

// ---------------------------------------------------------------------------
extern "C" void kernel_launch(void* const* d_in, const int* in_sizes, int n_in,
                              void* d_out, int out_size, void* d_ws, size_t ws_size,
                              hipStream_t stream) {
  const float* x      = (const float*)d_in[0];           // [n_nodes, 128]
  const long long* ei = (const long long*)d_in[1];       // [2, n_edges]
  const float* W      = (const float*)d_in[2];           // [128, 128]
  float* out          = (float*)d_out;

  const int n_nodes       = in_sizes[0] / N_FEAT;
  const long long n_edges = (long long)in_sizes[1] / 2;

  float* Q = (float*)d_ws;  // n_nodes*128*4 = 25.6 MB scratch

  // Stage 1: Q = x @ W^T  (WMMA f32)
  const int m_tiles = (n_nodes + 15) / 16;
  qgemm_wmma_f32<<<dim3(m_tiles), dim3(256), 0, stream>>>(x, W, Q, n_nodes);

  // Stage 2: per-edge dots (8 edges per 256-thread block, 1 wave per edge)
  const int eblocks = (int)((n_edges + 7) / 8);
  edge_dot<<<dim3(eblocks), dim3(256), 0, stream>>>(Q, ei, out, n_edges);
}